// Synthesizer_27479200760484
// MI455X (gfx1250) — compile-verified
//
#include <hip/hip_runtime.h>
#include <math.h>

typedef __attribute__((ext_vector_type(2))) float v2f;
typedef __attribute__((ext_vector_type(8))) float v8f;
typedef __attribute__((ext_vector_type(4))) unsigned int v4u;
typedef __attribute__((ext_vector_type(8))) int v8i;
typedef __attribute__((ext_vector_type(4))) int v4i;

#define B_TOK   4096
#define IN_DIM  1024
#define DIM     512
#define NNODES  8
#define SDIM    128
#define MAXOPS  4

#define KS    64    // gemm K-chunk (rows per TDM tile)
#define GSTR  72    // gemm LDS B-tile row stride (64 + 8 pad dwords)
#define KH    32    // hop K-chunk
#define HSTR  136   // hop LDS B-tile row stride (128 + 8 pad dwords)

// --- CDNA5 f32 WMMA: D(16x16) = A(16x4) * B(4x16) + C (codegen-verified) --
__device__ __forceinline__ v8f wmma4(v2f a, v2f b, v8f c) {
  return __builtin_amdgcn_wmma_f32_16x16x4_f32(false, a, false, b,
                                               (short)0, c, false, false);
}

__device__ __forceinline__ float gelu_tanh(float x) {
  const float k0 = 0.7978845608028654f;  // sqrt(2/pi)
  const float k1 = 0.044715f;
  return 0.5f * x * (1.0f + tanhf(k0 * (x + k1 * x * x * x)));
}

__device__ __forceinline__ unsigned lds_off_of(const void* p) {
  // generic pointer to __shared__: low 32 bits are the LDS byte address
  return (unsigned)(unsigned long long)(uintptr_t)p;
}

// --- B-operand pair loads: force ds_load_2addr_b32 to pair {x, x+STRIDE} so
// the result lands directly in an even-aligned WMMA operand pair (no repack
// v_movs). Mnemonic/format copied from compiler output. Manual s_wait_dscnt
// glue carries the values as "+v" so WMMA is data-ordered after the wait.
__device__ __forceinline__ v2f ds2_g0(unsigned a) {
  v2f r; asm volatile("ds_load_2addr_b32 %0, %1 offset0:0 offset1:72"
                      : "=v"(r) : "v"(a)); return r; }
__device__ __forceinline__ v2f ds2_g1(unsigned a) {
  v2f r; asm volatile("ds_load_2addr_b32 %0, %1 offset0:16 offset1:88"
                      : "=v"(r) : "v"(a)); return r; }
__device__ __forceinline__ v2f ds2_g2(unsigned a) {
  v2f r; asm volatile("ds_load_2addr_b32 %0, %1 offset0:32 offset1:104"
                      : "=v"(r) : "v"(a)); return r; }
__device__ __forceinline__ v2f ds2_g3(unsigned a) {
  v2f r; asm volatile("ds_load_2addr_b32 %0, %1 offset0:48 offset1:120"
                      : "=v"(r) : "v"(a)); return r; }
__device__ __forceinline__ v2f ds2_h0(unsigned a) {
  v2f r; asm volatile("ds_load_2addr_b32 %0, %1 offset0:0 offset1:136"
                      : "=v"(r) : "v"(a)); return r; }
__device__ __forceinline__ v2f ds2_h1(unsigned a) {
  v2f r; asm volatile("ds_load_2addr_b32 %0, %1 offset0:16 offset1:152"
                      : "=v"(r) : "v"(a)); return r; }
__device__ __forceinline__ void wait_ds4(v2f& b0, v2f& b1, v2f& b2, v2f& b3) {
  asm volatile("s_wait_dscnt 0x0" : "+v"(b0), "+v"(b1), "+v"(b2), "+v"(b3)); }
__device__ __forceinline__ void wait_ds2(v2f& b0, v2f& b1) {
  asm volatile("s_wait_dscnt 0x0" : "+v"(b0), "+v"(b1)); }

// =========================================================================
// Tensor Data Mover: 2D f32 tile (tile_h rows x tile_w dwords) from global
// row-major (row stride = row_stride elems) into LDS, with LDS row padding
// via D# pad fields (CDNA5 ISA). Wave-uniform args. 6-arg form:
//   (uint32x4 g0, int32x8 g1, int32x4, int32x4, int32x8, i32 cpol)
// =========================================================================
__device__ __forceinline__ void tdm_load_2d(unsigned lds_byte_off,
                                            const float* gaddr,
                                            unsigned tile_w, unsigned tile_h,
                                            unsigned row_stride,
                                            unsigned pad_iv, unsigned pad_am) {
  unsigned long long ga = (unsigned long long)(uintptr_t)gaddr;
  v4u g0;
  g0[0] = 1u;                                            // count=1, no gather
  g0[1] = lds_byte_off;                                  // lds_addr
  g0[2] = (unsigned)(ga & 0xFFFFFFFFu);                  // global_addr[31:0]
  g0[3] = (unsigned)((ga >> 32) & 0x01FFFFFFu) | (2u << 30); // [56:32]|type=2
  const unsigned d0 = (2u << 16)                         // data_size = 4B
                    | (1u << 20)                         // pad_enable
                    | (pad_iv << 22) | (pad_am << 25);
  const unsigned td0 = tile_w, td1 = tile_h;             // tensor dims == tile
  v8i g1;
  g1[0] = (int)d0;                                       // mask=0 (no cluster)
  g1[1] = (int)((td0 & 0xFFFFu) << 16);                  // tensor_dim0 lo
  g1[2] = (int)(((td0 >> 16) & 0xFFFFu) | ((td1 & 0xFFFFu) << 16));
  g1[3] = (int)(((td1 >> 16) & 0xFFFFu) | ((tile_w & 0xFFFFu) << 16));
  g1[4] = (int)(tile_h & 0xFFFFu);                       // tile_dim1, dim2=0
  g1[5] = (int)row_stride;                               // dim0_stride[31:0]
  g1[6] = 0;
  g1[7] = 0;
  v4i z4 = {0, 0, 0, 0};                                 // 2D: groups 2,3 off
  v8i z8 = {0, 0, 0, 0, 0, 0, 0, 0};
  __builtin_amdgcn_tensor_load_to_lds(g0, g1, z4, z4, z8, 0);
}

// =========================================================================
// f32 WMMA GEMM, TDM-staged double-buffered B tiles.
// Block = 256 thr (8 waves). Wave w: rows [bx*256 + w*32, +32) (2 row tiles),
// cols [by*64, +64) (4 frags). Per k-step: 2x global b64 (A), 4x
// ds_load_2addr_b32 (B operand pairs), 8 WMMA. gridDim.z batches slices.
// =========================================================================
__global__ void __launch_bounds__(256)
gemm_f32_wmma(const float* __restrict__ A, int lda,
              const float* __restrict__ W, int ldw, long wStrideZ,
              const float* __restrict__ bias,
              float* __restrict__ C, int ldc, long cStrideZ, int K) {
  __shared__ float tb[2][KS * GSTR];
  const int lane = threadIdx.x & 31;
  const int wave = threadIdx.x >> 5;
  const int m    = lane & 15;
  const int hi   = lane >> 4;
  const int row0 = blockIdx.x * 256 + wave * 32;
  const int col0 = blockIdx.y * 64;

  const float* Wz = W + (size_t)blockIdx.z * (size_t)wStrideZ + col0;
  float*       Cz = C + (size_t)blockIdx.z * (size_t)cStrideZ;
  const float* Ar0 = A + (size_t)(row0 + m) * lda;
  const float* Ar1 = A + (size_t)(row0 + 16 + m) * lda;

  const int nchunks = K / KS;
  if (threadIdx.x < 32)  // wave 0 drives the TDM
    tdm_load_2d(lds_off_of(&tb[0][0]), Wz, 64, KS, (unsigned)ldw, 5, 7);

  v8f acc[2][4];
  #pragma unroll
  for (int r = 0; r < 2; ++r)
    #pragma unroll
    for (int t = 0; t < 4; ++t)
      acc[r][t] = (v8f){0.f,0.f,0.f,0.f,0.f,0.f,0.f,0.f};

  for (int c = 0; c < nchunks; ++c) {
    if (threadIdx.x < 32) __builtin_amdgcn_s_wait_tensorcnt(0);
    __syncthreads();                       // tile c visible to all waves
    if (threadIdx.x < 32 && (c + 1) < nchunks)
      tdm_load_2d(lds_off_of(&tb[(c + 1) & 1][0]),
                  Wz + (size_t)(c + 1) * KS * ldw, 64, KS, (unsigned)ldw, 5, 7);

    const unsigned tA = lds_off_of(&tb[c & 1][0]) +
                        (unsigned)(((2 * hi) * GSTR + m) * 4);
    const float* A0 = Ar0 + c * KS + 2 * hi;
    const float* A1 = Ar1 + c * KS + 2 * hi;
    #pragma unroll
    for (int kk = 0; kk < KS; kk += 4) {
      const unsigned ak = tA + (unsigned)(kk * GSTR * 4);
      v2f b0 = ds2_g0(ak), b1 = ds2_g1(ak), b2 = ds2_g2(ak), b3 = ds2_g3(ak);
      v2f a0; a0[0] = A0[kk]; a0[1] = A0[kk + 1];
      v2f a1; a1[0] = A1[kk]; a1[1] = A1[kk + 1];
      wait_ds4(b0, b1, b2, b3);
      acc[0][0] = wmma4(a0, b0, acc[0][0]);
      acc[0][1] = wmma4(a0, b1, acc[0][1]);
      acc[0][2] = wmma4(a0, b2, acc[0][2]);
      acc[0][3] = wmma4(a0, b3, acc[0][3]);
      acc[1][0] = wmma4(a1, b0, acc[1][0]);
      acc[1][1] = wmma4(a1, b1, acc[1][1]);
      acc[1][2] = wmma4(a1, b2, acc[1][2]);
      acc[1][3] = wmma4(a1, b3, acc[1][3]);
    }
    // next-chunk TDM target is only overwritten after the *following*
    // start-of-chunk barrier, by which time all readers are done.
  }

  #pragma unroll
  for (int r = 0; r < 2; ++r)
    #pragma unroll
    for (int t = 0; t < 4; ++t) {
      const int col = col0 + t * 16 + m;
      const float bv = bias ? bias[col] : 0.0f;
      #pragma unroll
      for (int v = 0; v < 8; ++v)
        Cz[(size_t)(row0 + r * 16 + v + 8 * hi) * ldc + col] = acc[r][t][v] + bv;
    }
}

// =========================================================================
// sym_mean[b,s] = mean_n sym[b,n,s]
// =========================================================================
__global__ void sym_mean_kernel(const float* __restrict__ sym,
                                float* __restrict__ mean) {
  const int i = blockIdx.x * blockDim.x + threadIdx.x;
  if (i >= B_TOK * SDIM) return;
  const int b = i / SDIM, s = i % SDIM;
  const float* p = sym + (size_t)b * (NNODES * SDIM) + s;
  float acc = 0.f;
  #pragma unroll
  for (int n = 0; n < NNODES; ++n) acc += p[n * SDIM];
  mean[i] = acc * (1.0f / NNODES);
}

// =========================================================================
// Router: per-token logits + first-max argmax (as jnp.argmax).
// =========================================================================
__global__ void router_kernel(const float* __restrict__ z,
                              const float* __restrict__ symmean,
                              const float* __restrict__ rW,
                              const float* __restrict__ rb,
                              int* __restrict__ progI,
                              float* __restrict__ progF) {
  constexpr int NL = MAXOPS * (NNODES + 1);  // 36
  __shared__ float logits[NL];
  const int b = blockIdx.x;
  const int j = threadIdx.x;
  if (j < NL) {
    float acc = rb[j];
    const float* zb = z + (size_t)b * DIM;
    for (int d = 0; d < DIM; ++d) acc += zb[d] * rW[(size_t)d * NL + j];
    const float* sb = symmean + (size_t)b * SDIM;
    for (int s = 0; s < SDIM; ++s) acc += sb[s] * rW[(size_t)(DIM + s) * NL + j];
    logits[j] = acc;
  }
  __syncthreads();
  if (j < MAXOPS) {
    const float* L = logits + j * (NNODES + 1);
    int best = 0; float bv = L[0];
    #pragma unroll
    for (int t = 1; t < NNODES + 1; ++t)
      if (L[t] > bv) { bv = L[t]; best = t; }
    progI[b * MAXOPS + j] = best;
    progF[b * MAXOPS + j] = (float)best;
  }
}

// =========================================================================
// Deterministic counting sort of tokens by (hop, effective expert).
// =========================================================================
#define NCHK  64
#define CHKSZ (B_TOK / NCHK)   // 64 tokens per chunk
#define NBIN  (NNODES + 1)     // 9

__device__ __forceinline__ int eff_expert(const int* prog, int b, int h) {
  int still = 1;
  for (int hh = 0; hh <= h; ++hh) still &= (prog[b * MAXOPS + hh] != NNODES);
  return still ? prog[b * MAXOPS + h] : NNODES;
}

__global__ void route_count_kernel(const int* __restrict__ prog,
                                   int* __restrict__ cnts) {
  const int c = blockIdx.x, h = threadIdx.x;   // 64 blocks x 32 thr
  if (h >= MAXOPS) return;
  int cnt[NBIN];
  #pragma unroll
  for (int k = 0; k < NBIN; ++k) cnt[k] = 0;
  for (int i = 0; i < CHKSZ; ++i) cnt[eff_expert(prog, c * CHKSZ + i, h)]++;
  #pragma unroll
  for (int k = 0; k < NBIN; ++k) cnts[h * (NBIN * NCHK) + k * NCHK + c] = cnt[k];
}

__global__ void route_prefix_kernel(const int* __restrict__ cnts,
                                    int* __restrict__ base) {
  const int h = threadIdx.x;                   // 1 block x 32 thr
  if (h >= MAXOPS) return;
  int run = 0;
  for (int j = 0; j < NBIN * NCHK; ++j) {
    base[h * (NBIN * NCHK) + j] = run;
    run += cnts[h * (NBIN * NCHK) + j];
  }
}

__global__ void route_scatter_kernel(const int* __restrict__ prog,
                                     const int* __restrict__ base,
                                     int* __restrict__ perm) {
  const int c = blockIdx.x, h = threadIdx.x;   // 64 blocks x 32 thr
  if (h >= MAXOPS) return;
  int pos[NBIN];
  #pragma unroll
  for (int k = 0; k < NBIN; ++k) pos[k] = base[h * (NBIN * NCHK) + k * NCHK + c];
  for (int i = 0; i < CHKSZ; ++i) {
    const int b = c * CHKSZ + i;
    const int e = eff_expert(prog, b, h);
    perm[h * B_TOK + pos[e]++] = b;
  }
}

// =========================================================================
// One routing hop over expert-sorted tokens. Block = 128 thr (4 waves):
// 64 permuted rows x 128 cols. Wave w: all 4 row tiles, cols [c0+w*32,+32)
// (2 frags) -> 8 WMMA per k-step. Sorted tiles contain ~1 expert; the
// 512x128 expert panel is TDM-staged into double-buffered LDS.
// =========================================================================
__global__ void __launch_bounds__(128)
hop_kernel(const float* __restrict__ in, float* __restrict__ out,
           const int* __restrict__ prog, const int* __restrict__ perm,
           int hop, const float* __restrict__ opsW,
           const float* __restrict__ opsb) {
  __shared__ float tbB[2][KH * HSTR];
  __shared__ int s_row[64];
  __shared__ int s_exp[64];
  __shared__ int s_used;
  const int lane = threadIdx.x & 31;
  const int wave = threadIdx.x >> 5;
  const int m    = lane & 15;
  const int hi   = lane >> 4;
  const int r0   = blockIdx.x * 64;
  const int c0   = blockIdx.y * 128;

  if (threadIdx.x == 0) s_used = 0;
  __syncthreads();
  if (threadIdx.x < 64) {
    const int b = perm[hop * B_TOK + r0 + threadIdx.x];
    const int e = eff_expert(prog, b, hop);
    s_row[threadIdx.x] = b;
    s_exp[threadIdx.x] = e;
    if (e < NNODES) atomicOr(&s_used, 1 << e);
  }
  __syncthreads();

  // carry forward stopped rows (this block's column chunk)
  for (int i = threadIdx.x; i < 64 * 128; i += 128) {
    const int r = i >> 7, cc = i & 127;
    if (s_exp[r] == NNODES)
      out[(size_t)s_row[r] * DIM + c0 + cc] =
          in[(size_t)s_row[r] * DIM + c0 + cc];
  }

  const int used = s_used;
  if (!used) return;                       // uniform: whole tile stopped
  const float* Ar[4];
  #pragma unroll
  for (int rt = 0; rt < 4; ++rt)
    Ar[rt] = in + (size_t)s_row[rt * 16 + m] * DIM + 2 * hi;

  for (int e = 0; e < NNODES; ++e) {       // uniform branch -> EXEC all 1s
    if (!((used >> e) & 1)) continue;
    const float* We = opsW + (size_t)e * DIM * DIM + c0;
    constexpr int nchunks = DIM / KH;      // 16
    if (threadIdx.x < 32)
      tdm_load_2d(lds_off_of(&tbB[0][0]), We, 128, KH, DIM, 6, 7);

    v8f acc[4][2];
    #pragma unroll
    for (int rt = 0; rt < 4; ++rt)
      #pragma unroll
      for (int t = 0; t < 2; ++t)
        acc[rt][t] = (v8f){0.f,0.f,0.f,0.f,0.f,0.f,0.f,0.f};

    for (int c = 0; c < nchunks; ++c) {
      if (threadIdx.x < 32) __builtin_amdgcn_s_wait_tensorcnt(0);
      __syncthreads();
      if (threadIdx.x < 32 && (c + 1) < nchunks)
        tdm_load_2d(lds_off_of(&tbB[(c + 1) & 1][0]),
                    We + (size_t)(c + 1) * KH * DIM, 128, KH, DIM, 6, 7);

      const unsigned tA = lds_off_of(&tbB[c & 1][0]) +
                          (unsigned)(((2 * hi) * HSTR + wave * 32 + m) * 4);
      #pragma unroll
      for (int kk = 0; kk < KH; kk += 4) {
        const unsigned ak = tA + (unsigned)(kk * HSTR * 4);
        v2f b0 = ds2_h0(ak), b1 = ds2_h1(ak);
        v2f a[4];
        #pragma unroll
        for (int rt = 0; rt < 4; ++rt) {
          a[rt][0] = Ar[rt][c * KH + kk];
          a[rt][1] = Ar[rt][c * KH + kk + 1];
        }
        wait_ds2(b0, b1);
        #pragma unroll
        for (int rt = 0; rt < 4; ++rt) {
          acc[rt][0] = wmma4(a[rt], b0, acc[rt][0]);
          acc[rt][1] = wmma4(a[rt], b1, acc[rt][1]);
        }
      }
    }
    // epilogue: bias + gelu, write only rows routed to expert e
    #pragma unroll
    for (int rt = 0; rt < 4; ++rt)
      #pragma unroll
      for (int t = 0; t < 2; ++t) {
        const int col = c0 + wave * 32 + t * 16 + m;
        const float bb = opsb[e * DIM + col];
        #pragma unroll
        for (int v = 0; v < 8; ++v) {
          const int r = rt * 16 + v + 8 * hi;
          if (s_exp[r] == e)
            out[(size_t)s_row[r] * DIM + col] = gelu_tanh(acc[rt][t][v] + bb);
        }
      }
  }
}

// =========================================================================
extern "C" void kernel_launch(void* const* d_in, const int* in_sizes, int n_in,
                              void* d_out, int out_size, void* d_ws, size_t ws_size,
                              hipStream_t stream) {
  (void)in_sizes; (void)n_in; (void)out_size; (void)ws_size;
  const float* x        = (const float*)d_in[0];
  const float* W_in     = (const float*)d_in[1];
  const float* b_in     = (const float*)d_in[2];
  const float* ops_W    = (const float*)d_in[3];
  const float* ops_b    = (const float*)d_in[4];
  const float* sym_W    = (const float*)d_in[5];
  const float* router_W = (const float*)d_in[6];
  const float* router_b = (const float*)d_in[7];

  // d_out = out[B,D] ++ prog[B,MAXOPS] (float) ++ sym[B,N,S]
  float* outZ   = (float*)d_out;
  float* progF  = outZ  + (size_t)B_TOK * DIM;
  float* symOut = progF + (size_t)B_TOK * MAXOPS;

  // workspace
  float* z       = (float*)d_ws;
  float* buf     = z + (size_t)B_TOK * DIM;
  float* symmean = buf + (size_t)B_TOK * DIM;
  int*   progI   = (int*)(symmean + (size_t)B_TOK * SDIM);
  int*   cnts    = progI + (size_t)B_TOK * MAXOPS;          // [4][9][64]
  int*   basep   = cnts + MAXOPS * NBIN * NCHK;             // [4][9][64]
  int*   perm    = basep + MAXOPS * NBIN * NCHK;            // [4][B]

  // 1) z = x @ W_in + b_in
  gemm_f32_wmma<<<dim3(B_TOK / 256, DIM / 64, 1), 256, 0, stream>>>(
      x, IN_DIM, W_in, DIM, 0, b_in, z, DIM, 0, IN_DIM);

  // 2) sym[:,n,:] = z @ sym_W[n]
  gemm_f32_wmma<<<dim3(B_TOK / 256, SDIM / 64, NNODES), 256, 0, stream>>>(
      z, DIM, sym_W, SDIM, (long)DIM * SDIM, nullptr,
      symOut, NNODES * SDIM, SDIM, DIM);

  // 3) symmean
  sym_mean_kernel<<<(B_TOK * SDIM + 255) / 256, 256, 0, stream>>>(symOut, symmean);

  // 4) router -> prog
  router_kernel<<<B_TOK, 64, 0, stream>>>(z, symmean, router_W, router_b,
                                          progI, progF);

  // 5) deterministic counting sort of tokens by (hop, effective expert)
  route_count_kernel  <<<NCHK, 32, 0, stream>>>(progI, cnts);
  route_prefix_kernel <<<1,    32, 0, stream>>>(cnts, basep);
  route_scatter_kernel<<<NCHK, 32, 0, stream>>>(progI, basep, perm);

  // 6) 4 routing hops over sorted tokens; last hop writes d_out
  const dim3 hgrid(B_TOK / 64, DIM / 128);
  hop_kernel<<<hgrid, 128, 0, stream>>>(z,   buf,  progI, perm, 0, ops_W, ops_b);
  hop_kernel<<<hgrid, 128, 0, stream>>>(buf, z,    progI, perm, 1, ops_W, ops_b);
  hop_kernel<<<hgrid, 128, 0, stream>>>(z,   buf,  progI, perm, 2, ops_W, ops_b);
  hop_kernel<<<hgrid, 128, 0, stream>>>(buf, outZ, progI, perm, 3, ops_W, ops_b);
}